// ViewSaliencyLayer_33595234189759
// MI455X (gfx1250) — compile-verified
//
#include <hip/hip_runtime.h>

typedef __attribute__((ext_vector_type(2))) float v2f;
typedef __attribute__((ext_vector_type(8))) float v8f;
typedef int i32x4 __attribute__((vector_size(16)));

#define D_DIM 1024
#define N_DIM 128
#define KC 32                      // D-rows per LDS chunk
#define NCHUNK (D_DIM / KC)        // 32 chunks
#define VEC_PER_THREAD 4           // 4 x b128 async loads per thread per chunk
#define EPS 1e-10f

// ---- upper-triangular tile assignment: 36 tiles (tr<=tc) over 8 waves ----
constexpr int NTILE(int w) { return (w & 1) ? 4 : 5; }
constexpr int TR_TAB[8][5] = {
    {0,0,0,0,7}, {0,0,0,0,0}, {1,1,1,1,6}, {1,1,1,6,0},
    {2,2,2,2,5}, {2,2,5,5,0}, {3,3,3,3,3}, {4,4,4,4,0}};
constexpr int TC_TAB[8][5] = {
    {0,1,2,3,7}, {4,5,6,7,0}, {1,2,3,4,6}, {5,6,7,7,0},
    {2,3,4,5,5}, {6,7,6,7,0}, {3,4,5,6,7}, {4,5,6,7,0}};

constexpr int needed_mask(int w) {
    int m = 0;
    for (int t = 0; t < NTILE(w); ++t) {
        m |= 1 << TR_TAB[w][t];
        m |= 1 << TC_TAB[w][t];
    }
    return m;
}

// ---- 16-byte global->LDS copy: async path if available, sync fallback ----
__device__ __forceinline__ void copy16(const float* g, float* l) {
#if __has_builtin(__builtin_amdgcn_global_load_async_to_lds_b128)
    __builtin_amdgcn_global_load_async_to_lds_b128(
        (__attribute__((address_space(1))) i32x4*)g,
        (__attribute__((address_space(3))) i32x4*)l,
        0, 0);
#else
    float4 t = *(const float4*)g;
    *(float4*)l = t;
#endif
}

template <int N>
__device__ __forceinline__ void wait_async() {
#if __has_builtin(__builtin_amdgcn_global_load_async_to_lds_b128)
#if __has_builtin(__builtin_amdgcn_s_wait_asynccnt)
    __builtin_amdgcn_s_wait_asynccnt((short)N);
#else
    asm volatile("s_wait_asynccnt %0" ::"n"(N) : "memory");
#endif
#endif
}

__device__ __forceinline__ void issue_chunk(const float* __restrict__ xb,
                                            float* __restrict__ lds,
                                            int chunk, int tid) {
    const float* src = xb + (size_t)chunk * (KC * N_DIM);
#pragma unroll
    for (int i = 0; i < VEC_PER_THREAD; ++i) {
        int off = (tid + i * 256) * 4;   // float offset, 16B aligned
        copy16(src + off, lds + off);
    }
}

template <int W>
__device__ __forceinline__ void wave_compute(
    const float* __restrict__ xb, float* __restrict__ outb,
    float (&stage)[2][KC * N_DIM], float (&sqn)[N_DIM], float (&Srow)[N_DIM],
    float& total_s, int tid, int lane) {
    constexpr int NT = NTILE(W);
    constexpr int MASK = needed_mask(W);
    const int lg = lane >> 4;   // 0/1 half-wave group
    const int lc = lane & 15;

    v8f acc[NT];
#pragma unroll
    for (int t = 0; t < NT; ++t)
#pragma unroll
        for (int k = 0; k < 8; ++k) acc[t][k] = 0.0f;

    // ---------------- Phase 1: triangular Gram accumulation over D --------
    issue_chunk(xb, &stage[0][0], 0, tid);

#pragma unroll 1
    for (int c = 0; c < NCHUNK; ++c) {
        if (c + 1 < NCHUNK) {
            issue_chunk(xb, &stage[(c + 1) & 1][0], c + 1, tid);
            wait_async<VEC_PER_THREAD>();  // chunk c's loads (in-order) done
        } else {
            wait_async<0>();
        }
        __syncthreads();

        const float* bufp = &stage[c & 1][0];
#pragma unroll
        for (int dd = 0; dd < KC; dd += 4) {
            const int r0 = dd + 2 * lg;  // K rows r0, r0+1 (K = 2*lg + v)
            v2f blk[8];
#pragma unroll
            for (int bk = 0; bk < 8; ++bk) {
                if (MASK & (1 << bk)) {
                    blk[bk].x = bufp[r0 * N_DIM + 16 * bk + lc];
                    blk[bk].y = bufp[(r0 + 1) * N_DIM + 16 * bk + lc];
                }
            }
#pragma unroll
            for (int t = 0; t < NT; ++t)
                acc[t] = __builtin_amdgcn_wmma_f32_16x16x4_f32(
                    false, blk[TR_TAB[W][t]], false, blk[TC_TAB[W][t]],
                    (short)0, acc[t], false, false);
        }
        __syncthreads();
    }

    // ---------------- Phase 2: distances + symmetric row sums ------------
    // C/D layout: vgpr r, lane L -> row M = r + 8*(L>>4), col N = L&15.
#pragma unroll
    for (int t = 0; t < NT; ++t) {
        const int tr = TR_TAB[W][t], tc = TC_TAB[W][t];
        if (tr == tc) {  // extract squared norms from diagonal tiles
#pragma unroll
            for (int r = 0; r < 8; ++r) {
                int lrow = r + 8 * lg;
                if (lc == lrow) sqn[16 * tr + lrow] = acc[t][r];
            }
        }
    }
    __syncthreads();

#pragma unroll
    for (int t = 0; t < NT; ++t) {
        const int tr = TR_TAB[W][t], tc = TC_TAB[W][t];
        const int m  = 16 * tc + lc;
        const float sm = sqn[m];
        float colp = 0.0f;
#pragma unroll
        for (int r = 0; r < 8; ++r) {
            int n    = 16 * tr + r + 8 * lg;
            float g  = acc[t][r];
            float d2 = sqn[n] + sm - 2.0f * g;
            d2 = fmaxf(d2, 0.0f);
            float dist = (n != m && d2 > 0.0f) ? sqrtf(d2) : 0.0f;
            // reduce this row-slice across the 16 column-lanes
            float v = dist;
            v += __shfl_xor(v, 1, 32);
            v += __shfl_xor(v, 2, 32);
            v += __shfl_xor(v, 4, 32);
            v += __shfl_xor(v, 8, 32);
            if (lc == 0) atomicAdd(&Srow[n], v);
            if (tr != tc) colp += dist;  // symmetric contribution dist[m,n]
        }
        if (tr != tc) atomicAdd(&Srow[m], colp);
    }
    __syncthreads();

    if (tid == 0) {
        float tt = 0.0f;
        for (int n = 0; n < N_DIM; ++n) tt += Srow[n];
        total_s = tt;
    }
    __syncthreads();

    if (tid < N_DIM) outb[tid] = Srow[tid] / (total_s + EPS);
}

__global__ __launch_bounds__(256) void view_saliency_gram_kernel(
    const float* __restrict__ x, float* __restrict__ out) {
    __shared__ __align__(16) float stage[2][KC * N_DIM];  // 2 x 16 KB
    __shared__ float sqn[N_DIM];
    __shared__ float Srow[N_DIM];
    __shared__ float total_s;

    const int tid  = threadIdx.x;
    const int lane = tid & 31;
    const int wave = tid >> 5;  // 0..7
    const int b    = blockIdx.x;

    const float* xb   = x + (size_t)b * (D_DIM * N_DIM);
    float*       outb = out + (size_t)b * N_DIM;

    if (tid < N_DIM) Srow[tid] = 0.0f;

    // Each case runs an identical barrier schedule (wave-uniform divergence).
    switch (wave) {
        case 0: wave_compute<0>(xb, outb, stage, sqn, Srow, total_s, tid, lane); break;
        case 1: wave_compute<1>(xb, outb, stage, sqn, Srow, total_s, tid, lane); break;
        case 2: wave_compute<2>(xb, outb, stage, sqn, Srow, total_s, tid, lane); break;
        case 3: wave_compute<3>(xb, outb, stage, sqn, Srow, total_s, tid, lane); break;
        case 4: wave_compute<4>(xb, outb, stage, sqn, Srow, total_s, tid, lane); break;
        case 5: wave_compute<5>(xb, outb, stage, sqn, Srow, total_s, tid, lane); break;
        case 6: wave_compute<6>(xb, outb, stage, sqn, Srow, total_s, tid, lane); break;
        default: wave_compute<7>(xb, outb, stage, sqn, Srow, total_s, tid, lane); break;
    }
}

extern "C" void kernel_launch(void* const* d_in, const int* in_sizes, int n_in,
                              void* d_out, int out_size, void* d_ws, size_t ws_size,
                              hipStream_t stream) {
    const float* x = (const float*)d_in[0];
    float* out = (float*)d_out;
    int batches = in_sizes[0] / (D_DIM * N_DIM);  // 256
    view_saliency_gram_kernel<<<dim3(batches), dim3(256), 0, stream>>>(x, out);
}